// GNLoss_87643102642329
// MI455X (gfx1250) — compile-verified
//
#include <hip/hip_runtime.h>
#include <hip/hip_bf16.h>
#include <math.h>

// ---------------------------------------------------------------------------
// GN loss on gfx1250: per-point 3x64 Gram via chained V_WMMA_F32_16X16X4_F32.
// ---------------------------------------------------------------------------

typedef __attribute__((ext_vector_type(2))) float v2f;
typedef __attribute__((ext_vector_type(8))) float v8f;

#define NTASK     8192          // 4 levels * B(4) * N(512)
#define NWAVE_B   1640          // ceil(8192/5) rounded up to 8 waves/block * 205 blocks
#define NPART     (NTASK + NWAVE_B)
#define INV_N     (1.0f / 512.0f)
#define LAMDA_F   0.003f
#define EPS_F     1e-9f
// 4 levels * LAMDA * B*N * log(2*pi)
#define E2_CONST  ((float)(4.0 * 0.003 * 2048.0 * 1.8378770664093454))

__device__ __forceinline__ const float* sel_lvl(int i, const float* p0, const float* p1,
                                                const float* p2, const float* p3) {
  return (i == 0) ? p0 : (i == 1) ? p1 : (i == 2) ? p2 : p3;
}

// D[M][N] of the 16x16 f32 accumulator: M<8 -> vgpr M, lane N ; M>=8 -> vgpr M-8, lane N+16.
__device__ __forceinline__ float d_elem(const v8f& acc, int M, int N) {
  int lane = (M < 8) ? N : (N + 16);
  int e = M & 7;
  float v;
  switch (e) {
    case 0: v = acc[0]; break;
    case 1: v = acc[1]; break;
    case 2: v = acc[2]; break;
    case 3: v = acc[3]; break;
    case 4: v = acc[4]; break;
    case 5: v = acc[5]; break;
    case 6: v = acc[6]; break;
    default: v = acc[7]; break;
  }
  return __shfl(v, lane, 32);
}

// ---------------------------------------------------------------------------
// Kernel 1: contrastive negative loss. One wave per task (level,b,n).
// ---------------------------------------------------------------------------
__global__ void neg_kernel(const float* __restrict__ fa0, const float* __restrict__ fa1,
                           const float* __restrict__ fa2, const float* __restrict__ fa3,
                           const float* __restrict__ fb0, const float* __restrict__ fb1,
                           const float* __restrict__ fb2, const float* __restrict__ fb3,
                           const int* __restrict__ na, const int* __restrict__ nb,
                           float* __restrict__ ws) {
  int gtid = blockIdx.x * blockDim.x + threadIdx.x;
  int wid  = gtid >> 5;
  int lane = gtid & 31;
  if (wid >= NTASK) return;

  int i   = wid >> 11;        // level 0..3
  int rem = wid & 2047;
  int b   = rem >> 9;
  int n   = rem & 511;
  int shift = 3 - i;
  int s   = 32 << i;
  long ss = (long)s * s;

  int idx = (((b << 9) + n) << 1);
  int xa = na[idx]     >> shift;
  int ya = na[idx + 1] >> shift;
  int xb = nb[idx]     >> shift;
  int yb = nb[idx + 1] >> shift;

  const float* FA = sel_lvl(i, fa0, fa1, fa2, fa3);
  const float* FB = sel_lvl(i, fb0, fb1, fb2, fb3);
  long base = (long)b * 64 * ss;
  const float* pa = FA + base + (long)ya * s + xa;
  const float* pb = FB + base + (long)yb * s + xb;

  // 64 channels, 2 per lane
  float d0 = pa[(long)lane * ss]        - pb[(long)lane * ss];
  float d1 = pa[(long)(lane + 32) * ss] - pb[(long)(lane + 32) * ss];
  float m0 = fmaxf(1.0f - d0, 0.0f);
  float m1 = fmaxf(1.0f - d1, 0.0f);
  float sum = m0 * m0 + m1 * m1;
  #pragma unroll
  for (int off = 16; off > 0; off >>= 1)
    sum += __shfl_xor(sum, off, 32);
  if (lane == 0) ws[wid] = sum * INV_N;
}

// ---------------------------------------------------------------------------
// Kernel 2: loss_pos + GN terms. One wave handles 5 point-tasks.
// Rows 0..14 of the 16x4 A-tile are [Jx;Jy;r] of 5 points; G = A*A^T by WMMA.
// A operand f32 16x4 layout: lane m (m<16): V0=K0,V1=K1 ; lane m+16: V0=K2,V1=K3.
// B = A^T is register-identical, so the same v2f feeds both operands.
// ---------------------------------------------------------------------------
__global__ void gram_wmma_kernel(const float* __restrict__ fa0, const float* __restrict__ fa1,
                                 const float* __restrict__ fa2, const float* __restrict__ fa3,
                                 const float* __restrict__ fb0, const float* __restrict__ fb1,
                                 const float* __restrict__ fb2, const float* __restrict__ fb3,
                                 const int* __restrict__ ka, const int* __restrict__ kb,
                                 float* __restrict__ ws) {
  int gtid  = blockIdx.x * blockDim.x + threadIdx.x;
  int wid   = gtid >> 5;
  int lane  = gtid & 31;
  int rho   = lane & 15;     // row of A (and column of B)
  int khalf = lane >> 4;     // 0: K=0,1  1: K=2,3

  int p = rho / 3;           // point in tile (0..4), rho==15 -> 5 (dead row)
  int v = rho - 3 * p;       // 0=Jx 1=Jy 2=r
  int t = wid * 5 + p;
  bool valid = (rho < 15) && (t < NTASK);
  int tc = valid ? t : 0;

  int i   = tc >> 11;
  int rem = tc & 2047;
  int b   = rem >> 9;
  int n   = rem & 511;
  int shift = 3 - i;
  int s   = 32 << i;
  long ss = (long)s * s;

  int idx = (((b << 9) + n) << 1);
  int xa = ka[idx]     >> shift;
  int ya = ka[idx + 1] >> shift;
  int xb = kb[idx]     >> shift;
  int yb = kb[idx + 1] >> shift;

  const float* FA = sel_lvl(i, fa0, fa1, fa2, fa3);
  const float* FB = sel_lvl(i, fb0, fb1, fb2, fb3);
  long base = (long)b * 64 * ss;

  int xp = min(xb + 1, s - 1), xm = max(xb - 1, 0);
  int yp = min(yb + 1, s - 1), ym = max(yb - 1, 0);

  long o1 = (v == 0) ? ((long)yb * s + xp)
          : (v == 1) ? ((long)yp * s + xb)
                     : ((long)yb * s + xb);
  long o2 = (v == 0) ? ((long)yb * s + xm)
          : (v == 1) ? ((long)ym * s + xb)
                     : ((long)ya * s + xa);
  const float* P1 = FB + base + o1;
  const float* P2 = ((v == 2) ? FA : FB) + base + o2;

  int coord = (v == 0) ? xb : yb;
  float sc = (v == 2) ? 1.0f : ((coord == 0 || coord == s - 1) ? 1.0f : 0.5f);
  if (!valid) sc = 0.0f;   // per-lane zeroing; EXEC stays all-ones for WMMA

  // Gather this lane's half-row: channels 4k+2*khalf, 4k+2*khalf+1 for k=0..15
  float A0[16], A1[16];
  #pragma unroll
  for (int k = 0; k < 16; ++k) {
    long c0 = 4 * k + 2 * khalf;
    A0[k] = (P1[c0 * ss]       - P2[c0 * ss])       * sc;
    A1[k] = (P1[(c0 + 1) * ss] - P2[(c0 + 1) * ss]) * sc;
  }

  v8f acc = {0.f, 0.f, 0.f, 0.f, 0.f, 0.f, 0.f, 0.f};
  #pragma unroll
  for (int k = 0; k < 16; ++k) {
    v2f a;
    a[0] = A0[k];
    a[1] = A1[k];
    acc = __builtin_amdgcn_wmma_f32_16x16x4_f32(
        /*neg_a=*/false, a, /*neg_b=*/false, a,
        /*c_mod=*/(short)0, acc, /*reuse_a=*/false, /*reuse_b=*/false);
  }

  // Extract 3x3 block-diagonal Grams and do the per-point scalar math.
  float wave_sum = 0.0f;
  #pragma unroll
  for (int pp = 0; pp < 5; ++pp) {
    const int m0 = 3 * pp;
    float G00 = d_elem(acc, m0,     m0);
    float G01 = d_elem(acc, m0,     m0 + 1);
    float G11 = d_elem(acc, m0 + 1, m0 + 1);
    float G02 = d_elem(acc, m0,     m0 + 2);
    float G12 = d_elem(acc, m0 + 1, m0 + 2);
    float G22 = d_elem(acc, m0 + 2, m0 + 2);
    if (wid * 5 + pp < NTASK) {   // wave-uniform
      float a_ = G00 + EPS_F;
      float c_ = G11 + EPS_F;
      float bb = G01;
      float det = a_ * c_ - bb * bb;
      float e1 = 0.5f * (c_ * G02 * G02 - 2.0f * bb * G02 * G12 + a_ * G12 * G12) / det;
      wave_sum += G22 * INV_N + LAMDA_F * (e1 - 0.5f * logf(det));
    }
  }
  if (lane == 0) ws[wid] = wave_sum;
}

// ---------------------------------------------------------------------------
// Kernel 3: deterministic fixed-order reduction of all partials.
// ---------------------------------------------------------------------------
__global__ void reduce_kernel(const float* __restrict__ ws, float* __restrict__ out) {
  __shared__ float sm[256];
  float s = 0.0f;
  for (int idx = threadIdx.x; idx < NPART; idx += 256) s += ws[idx];
  sm[threadIdx.x] = s;
  __syncthreads();
  #pragma unroll
  for (int off = 128; off > 0; off >>= 1) {
    if ((int)threadIdx.x < off) sm[threadIdx.x] += sm[threadIdx.x + off];
    __syncthreads();
  }
  if (threadIdx.x == 0) out[0] = sm[0] + E2_CONST;
}

// ---------------------------------------------------------------------------
extern "C" void kernel_launch(void* const* d_in, const int* in_sizes, int n_in,
                              void* d_out, int out_size, void* d_ws, size_t ws_size,
                              hipStream_t stream) {
  const float* fa0 = (const float*)d_in[0];
  const float* fa1 = (const float*)d_in[1];
  const float* fa2 = (const float*)d_in[2];
  const float* fa3 = (const float*)d_in[3];
  const float* fb0 = (const float*)d_in[4];
  const float* fb1 = (const float*)d_in[5];
  const float* fb2 = (const float*)d_in[6];
  const float* fb3 = (const float*)d_in[7];
  const int* ka = (const int*)d_in[8];
  const int* kb = (const int*)d_in[9];
  const int* na = (const int*)d_in[10];
  const int* nb = (const int*)d_in[11];

  float* wsA = (float*)d_ws;          // NTASK partials (neg loss)
  float* wsB = wsA + NTASK;           // NWAVE_B partials (pos + GN)

  // 8192 waves, 8 waves (256 threads) per block
  neg_kernel<<<NTASK / 8, 256, 0, stream>>>(fa0, fa1, fa2, fa3, fb0, fb1, fb2, fb3,
                                            na, nb, wsA);
  // 1640 waves = 205 blocks * 8 waves
  gram_wmma_kernel<<<NWAVE_B / 8, 256, 0, stream>>>(fa0, fa1, fa2, fa3, fb0, fb1, fb2, fb3,
                                                    ka, kb, wsB);
  reduce_kernel<<<1, 256, 0, stream>>>(wsA, (float*)d_out);
}